// GraphConvolution_37160057045703
// MI455X (gfx1250) — compile-verified
//
#include <hip/hip_runtime.h>
#include <hip/hip_bf16.h>

typedef __attribute__((ext_vector_type(2))) float v2f;
typedef __attribute__((ext_vector_type(8))) float v8f;

#define F_IN  128
#define F_OUT 64

// ---------------------------------------------------------------------------
// Kernel 1: xw = x @ W using fp32 WMMA (V_WMMA_F32_16X16X4_F32).
// Block = 256 threads = 8 waves, arranged as 2 (rows) x 4 (cols) tiles of
// 16x16 -> each block covers 32 rows x 64 cols of the output.
// ---------------------------------------------------------------------------
__global__ __launch_bounds__(256)
void gcn_xw_wmma_kernel(const float* __restrict__ x,
                        const float* __restrict__ W,
                        float* __restrict__ xw,
                        int n_nodes) {
    const int lane = threadIdx.x & 31;
    const int wave = threadIdx.x >> 5;       // 0..7
    const int wm   = wave >> 2;              // 0..1  (16-row tile within block)
    const int wn   = wave & 3;               // 0..3  (16-col tile)

    const int row0 = blockIdx.x * 32 + wm * 16;
    const int col0 = wn * 16;

    const int n     = lane & 15;             // A row / B,C,D column
    const int khalf = (lane >> 4) << 1;      // 0 for lanes 0-15, 2 for 16-31

    // Clamp the A-row pointer so out-of-range waves still execute the WMMA
    // loop with EXEC all-ones (WMMA requires full EXEC); stores are guarded.
    int arow_idx = row0 + n;
    if (arow_idx >= n_nodes) arow_idx = n_nodes - 1;
    const float* __restrict__ arow = x + (size_t)arow_idx * F_IN;

    v8f acc = {};
    #pragma unroll 8
    for (int k0 = 0; k0 < F_IN; k0 += 4) {
        // A: 16x4 tile, lane holds A[m][khalf], A[m][khalf+1] (contiguous b64)
        v2f a = *(const v2f*)(arow + k0 + khalf);
        // B: 4x16 tile, lane holds W[k0+khalf][col0+n], W[k0+khalf+1][col0+n]
        v2f bm;
        bm.x = W[(size_t)(k0 + khalf)     * F_OUT + col0 + n];
        bm.y = W[(size_t)(k0 + khalf + 1) * F_OUT + col0 + n];
        // args: (neg_a, A, neg_b, B, c_mod, C, reuse_a, reuse_b)
        acc = __builtin_amdgcn_wmma_f32_16x16x4_f32(
            false, a, false, bm, (short)0, acc, false, false);
    }

    // C/D layout: VGPR v holds row (v)   for lanes 0-15  (N = lane)
    //                         row (v+8) for lanes 16-31 (N = lane-16)
    const int moff = (lane >> 4) << 3;       // 0 or 8
    float* __restrict__ crow = xw + (size_t)(row0 + moff) * F_OUT + col0 + n;

    if (row0 + 16 <= n_nodes) {
        // Fast path (always taken when n_nodes % 32 == 0): unguarded stores,
        // constant immediate offsets from a single base.
        #pragma unroll
        for (int v = 0; v < 8; ++v) {
            crow[(size_t)v * F_OUT] = acc[v];
        }
    } else {
        #pragma unroll
        for (int v = 0; v < 8; ++v) {
            if (row0 + moff + v < n_nodes) {
                crow[(size_t)v * F_OUT] = acc[v];
            }
        }
    }
}

// ---------------------------------------------------------------------------
// Kernel 2: out[i][j] = b[j]  (bias initialization; aggregation adds on top)
// ---------------------------------------------------------------------------
__global__ __launch_bounds__(256)
void gcn_bias_init_kernel(const float* __restrict__ b,
                          float* __restrict__ out,
                          int total) {
    int i = blockIdx.x * blockDim.x + threadIdx.x;
    if (i < total) out[i] = b[i & (F_OUT - 1)];
}

// ---------------------------------------------------------------------------
// Kernel 3: edge scatter. 16 threads per edge, float4 gather from xw (L2-
// resident, 25.6 MB), 4 return-less fp32 atomic adds into out (L2 atomics).
// ---------------------------------------------------------------------------
__global__ __launch_bounds__(256)
void gcn_edge_scatter_kernel(const int* __restrict__ adj_row,
                             const int* __restrict__ adj_col,
                             const float* __restrict__ adj_val,
                             const float* __restrict__ xw,
                             float* __restrict__ out,
                             int n_edges) {
    int t = blockIdx.x * blockDim.x + threadIdx.x;
    int e = t >> 4;                          // edge index (16 threads/edge)
    if (e >= n_edges) return;
    int f = (t & 15) << 2;                   // feature offset 0,4,...,60

    int   r = adj_row[e];
    int   c = adj_col[e];
    float v = adj_val[e];

    const float4 d = *(const float4*)(xw + (size_t)c * F_OUT + f);
    float* dst = out + (size_t)r * F_OUT + f;

    unsafeAtomicAdd(dst + 0, v * d.x);       // -> global_atomic_add_f32 (no rtn)
    unsafeAtomicAdd(dst + 1, v * d.y);
    unsafeAtomicAdd(dst + 2, v * d.z);
    unsafeAtomicAdd(dst + 3, v * d.w);
}

// ---------------------------------------------------------------------------
extern "C" void kernel_launch(void* const* d_in, const int* in_sizes, int n_in,
                              void* d_out, int out_size, void* d_ws, size_t ws_size,
                              hipStream_t stream) {
    const float* x       = (const float*)d_in[0];
    const int*   adj_row = (const int*)  d_in[1];
    const int*   adj_col = (const int*)  d_in[2];
    const float* adj_val = (const float*)d_in[3];
    const float* W       = (const float*)d_in[4];
    const float* b       = (const float*)d_in[5];
    float*       out     = (float*)d_out;
    float*       xw      = (float*)d_ws;     // n_nodes * F_OUT floats (25.6 MB)

    const int n_nodes = in_sizes[0] / F_IN;
    const int n_edges = in_sizes[3];          // adj_val element count

    // Phase 1: dense projection via fp32 WMMA
    {
        dim3 grid((n_nodes + 31) / 32);
        gcn_xw_wmma_kernel<<<grid, 256, 0, stream>>>(x, W, xw, n_nodes);
    }
    // Phase 2: out = broadcast(b)
    {
        int total = n_nodes * F_OUT;
        dim3 grid((total + 255) / 256);
        gcn_bias_init_kernel<<<grid, 256, 0, stream>>>(b, out, total);
    }
    // Phase 3: out[r] += val * xw[c] over edges (fp32 L2 atomics)
    {
        long long threads = (long long)n_edges * 16;
        dim3 grid((unsigned)((threads + 255) / 256));
        gcn_edge_scatter_kernel<<<grid, 256, 0, stream>>>(
            adj_row, adj_col, adj_val, xw, out, n_edges);
    }
}